// GATv2_26912265077120
// MI455X (gfx1250) — compile-verified
//
#include <hip/hip_runtime.h>
#include <hip/hip_bf16.h>

typedef __attribute__((ext_vector_type(2))) float v2f;
typedef __attribute__((ext_vector_type(8))) float v8f;
typedef int v4i __attribute__((__vector_size__(16)));

static constexpr int NB = 2;
static constexpr int NN = 1024;

// ---- gfx1250 async global->LDS path (guarded; falls back to sync staging) ----
#if defined(__has_builtin)
#if __has_builtin(__builtin_amdgcn_global_load_async_to_lds_b128) && \
    __has_builtin(__builtin_amdgcn_global_load_async_to_lds_b32) &&  \
    __has_builtin(__builtin_amdgcn_s_wait_asynccnt)
#define USE_ASYNC_LDS 1
#endif
#endif
#ifndef USE_ASYNC_LDS
#define USE_ASYNC_LDS 0
#endif

#if USE_ASYNC_LDS
__device__ __forceinline__ void async_g2l_b128(const void* g, void* l) {
  __builtin_amdgcn_global_load_async_to_lds_b128(
      (__attribute__((address_space(1))) v4i*)(g),
      (__attribute__((address_space(3))) v4i*)(l), 0, 0);
}
__device__ __forceinline__ void async_g2l_b32(const void* g, void* l) {
  __builtin_amdgcn_global_load_async_to_lds_b32(
      (__attribute__((address_space(1))) int*)(g),
      (__attribute__((address_space(3))) int*)(l), 0, 0);
}
#endif

// ---------------------------------------------------------------------------
// GEMM: G[rows, NVALID] = X[rows, DK] @ W[NVALID, DK]^T  via V_WMMA_F32_16X16X4_F32
// 1 wave / block, 16 rows / block, NCOLS (multiple of 16) output columns.
// ---------------------------------------------------------------------------
template <int DK, int NCOLS, int NVALID>
__global__ __launch_bounds__(32) void gemm_xwt_wmma(const float* __restrict__ X,
                                                    const float* __restrict__ W,
                                                    float* __restrict__ G) {
  const int lane = threadIdx.x;
  const int hi   = lane >> 4;   // half-wave: K offset 0 or 2
  const int lm   = lane & 15;
  const long row0 = (long)blockIdx.x * 16;
  constexpr int NT = NCOLS / 16;

  v8f acc[NT];
#pragma unroll
  for (int t = 0; t < NT; ++t) acc[t] = v8f{};

  const float* xrow = X + (row0 + lm) * DK;
#pragma unroll 4
  for (int k0 = 0; k0 < DK; k0 += 4) {
    const int kb = k0 + 2 * hi;
    v2f a;
    a.x = xrow[kb];
    a.y = xrow[kb + 1];
#pragma unroll
    for (int t = 0; t < NT; ++t) {
      const int n = t * 16 + lm;      // output column (B-matrix N index)
      v2f bf;
      if (n < NVALID) {               // W^T[k][n] = W[n*DK+k]
        bf.x = W[n * DK + kb];
        bf.y = W[n * DK + kb + 1];
      } else {
        bf.x = 0.f; bf.y = 0.f;
      }
      acc[t] = __builtin_amdgcn_wmma_f32_16x16x4_f32(
          false, a, false, bf, (short)0, acc[t], false, false);
    }
  }
#pragma unroll
  for (int t = 0; t < NT; ++t) {
    const int n = t * 16 + lm;
    if (n < NVALID) {
#pragma unroll
      for (int r = 0; r < 8; ++r)
        G[(row0 + r + 8 * hi) * NVALID + n] = acc[t][r];
    }
  }
}

// ---------------------------------------------------------------------------
// GATv2 attention layer (flash-style online softmax + WMMA aggregation).
// HF = H*8 features; one wave per head; 16-row i-tile per block; j chunks of 16.
// Double-buffered async global->LDS staging of the g-tile and adjacency tile.
// ---------------------------------------------------------------------------
template <int HF, bool DO_ELU>
__global__ __launch_bounds__(32 * (HF / 8)) void gatv2_attn(
    const float* __restrict__ Gm,   // [NB*NN, HF] projected features
    const int* __restrict__ Adj,    // [NB, NN, NN]
    const float* __restrict__ Avec, // [8] attention vector (shared across heads)
    float* __restrict__ Out) {      // [NB*NN, HF]
  constexpr int F = 8;
  constexpr int H = HF / 8;
  constexpr int JC = 16;            // j-chunk
  constexpr int NCHUNK = NN / JC;

  __shared__ __align__(16) float gtile[2][JC * HF];
  __shared__ __align__(16) int   adjtile[2][16 * JC];

  const int tid  = threadIdx.x;
  const int lane = tid & 31;
  const int w    = tid >> 5;      // head
  const int hi   = lane >> 4;
  const int lm   = lane & 15;

  constexpr int TPB = NN / 16;
  const int b  = blockIdx.x / TPB;
  const int i0 = (blockIdx.x % TPB) * 16;

  float avec[F];
#pragma unroll
  for (int f = 0; f < F; ++f) avec[f] = Avec[f];

  // g_i for this head's i-tile: lane row = lm (duplicated across halves)
  float gi[F];
  {
    const float* p = Gm + ((long)b * NN + i0 + lm) * HF + w * F;
#pragma unroll
    for (int f = 0; f < F; ++f) gi[f] = p[f];
  }

  float m = -INFINITY;  // running row max  (row = lm)
  float l = 0.f;        // running row sum-of-exp
  v8f acc = v8f{};      // C fragment: rows r+8*hi, col lm (cols 0..7 valid)

  const long adjbase  = (long)b * NN * NN + (long)i0 * NN;
  const float* gjbase = Gm + (long)b * NN * HF;

  // ---- one chunk of the online-softmax + WMMA pipeline -------------------
  auto compute_chunk = [&](const float* gt, const int* at) {
    float e[8];
    float cmax = -INFINITY;
#pragma unroll
    for (int kk = 0; kk < 4; ++kk) {
#pragma unroll
      for (int u = 0; u < 2; ++u) {
        const int jloc = 4 * kk + 2 * hi + u;
        const float* gj = &gt[jloc * HF + w * F];
        float s = 0.f;
#pragma unroll
        for (int f = 0; f < F; ++f) {
          float t0 = gi[f] + gj[f];
          t0 = (t0 >= 0.f) ? t0 : 0.2f * t0;  // LeakyReLU(0.2)
          s = fmaf(avec[f], t0, s);
        }
        if (at[lm * 16 + jloc] == 0) s = -INFINITY;
        e[kk * 2 + u] = s;
        cmax = fmaxf(cmax, s);
      }
    }
    cmax = fmaxf(cmax, __shfl_xor(cmax, 16, 32));  // combine halves (same row)
    const float mnew  = fmaxf(m, cmax);
    const float mstar = (mnew == -INFINITY) ? 0.f : mnew;  // exp(-inf - -inf) guard
    const float scale = __expf(m - mstar);
    m = mnew;

    float psum = 0.f;
    v2f pf[4];
#pragma unroll
    for (int kk = 0; kk < 4; ++kk) {
      const float p0 = __expf(e[kk * 2 + 0] - mstar);
      const float p1 = __expf(e[kk * 2 + 1] - mstar);
      psum += p0 + p1;
      pf[kk].x = p0;
      pf[kk].y = p1;
    }
    psum += __shfl_xor(psum, 16, 32);
    l = l * scale + psum;

#pragma unroll
    for (int r = 0; r < 8; ++r) acc[r] *= __shfl(scale, r + 8 * hi, 32);

#pragma unroll
    for (int kk = 0; kk < 4; ++kk) {
      v2f bf;
      if (lm < F) {
        bf.x = gt[(4 * kk + 2 * hi + 0) * HF + w * F + lm];
        bf.y = gt[(4 * kk + 2 * hi + 1) * HF + w * F + lm];
      } else {
        bf.x = 0.f; bf.y = 0.f;
      }
      acc = __builtin_amdgcn_wmma_f32_16x16x4_f32(
          false, pf[kk], false, bf, (short)0, acc, false, false);
    }
  };

#if USE_ASYNC_LDS
  // async ops issued per thread per chunk: 1x b128 (g) + adj portion
  constexpr int KOPS = (H == 1) ? 3 : 2;
  auto issue_chunk = [&](int j0, int buf) {
    {  // g tile: 16*HF floats total -> exactly one b128 per thread
      const int fi = tid * 4;
      const int jj = fi / HF;
      const int ff = fi % HF;
      async_g2l_b128(gjbase + (long)(j0 + jj) * HF + ff, &gtile[buf][fi]);
    }
    if constexpr (H == 8) {  // 256 ints over 256 threads -> one b32 each
      const int idx = tid;
      async_g2l_b32(Adj + adjbase + (long)(idx >> 4) * NN + (j0 + (idx & 15)),
                    &adjtile[buf][idx]);
    } else {                 // 256 ints over 32 threads -> 8 ints = two b128 each
      const int idx = tid * 8;
      const int* src = Adj + adjbase + (long)(idx >> 4) * NN + (j0 + (idx & 15));
      async_g2l_b128(src, &adjtile[buf][idx]);
      async_g2l_b128(src + 4, &adjtile[buf][idx + 4]);
    }
  };

  issue_chunk(0, 0);
  for (int c = 0; c < NCHUNK; ++c) {
    __syncthreads();  // all waves done computing chunk c-1 -> its buffer is free
    if (c + 1 < NCHUNK) {
      issue_chunk((c + 1) * JC, (c + 1) & 1);
      __builtin_amdgcn_s_wait_asynccnt(KOPS);  // chunk c's DMAs (this wave) landed
    } else {
      __builtin_amdgcn_s_wait_asynccnt(0);
    }
    __syncthreads();  // chunk c staged block-wide
    compute_chunk(gtile[c & 1], adjtile[c & 1]);
  }
#else
  for (int j0 = 0; j0 < NN; j0 += JC) {
    __syncthreads();
    {  // sync staging: one float4 per thread
      const int fi = tid * 4;
      const int jj = fi / HF;
      const int ff = fi % HF;
      *(float4*)(&gtile[0][fi]) =
          *(const float4*)(gjbase + (long)(j0 + jj) * HF + ff);
    }
    {
      constexpr int API = 256 / (32 * H);
#pragma unroll
      for (int u = 0; u < API; ++u) {
        const int idx = tid * API + u;
        adjtile[0][idx] =
            Adj[adjbase + (long)(idx >> 4) * NN + (j0 + (idx & 15))];
      }
    }
    __syncthreads();
    if (j0 + JC < NN)
      __builtin_prefetch(gjbase + (long)(j0 + JC) * HF, 0, 0);
    compute_chunk(gtile[0], adjtile[0]);
  }
#endif

  // epilogue: out[i, h, f] = acc / l, optional ELU
  if (lm < F) {
#pragma unroll
    for (int r = 0; r < 8; ++r) {
      const float lr = __shfl(l, r + 8 * hi, 32);
      float v = acc[r] / lr;
      if (DO_ELU) v = (v > 0.f) ? v : (__expf(v) - 1.f);
      Out[((long)b * NN + i0 + r + 8 * hi) * HF + w * F + lm] = v;
    }
  }
}

// ---------------------------------------------------------------------------
extern "C" void kernel_launch(void* const* d_in, const int* in_sizes, int n_in,
                              void* d_out, int out_size, void* d_ws, size_t ws_size,
                              hipStream_t stream) {
  (void)in_sizes; (void)n_in; (void)out_size; (void)ws_size;
  const float* x   = (const float*)d_in[0];  // [2,1024,128]
  const int*   adj = (const int*)d_in[1];    // [2,1024,1024]
  const float* W1  = (const float*)d_in[2];  // [64,128]
  const float* a1  = (const float*)d_in[3];  // [8]
  const float* W2  = (const float*)d_in[4];  // [8,64]
  const float* a2  = (const float*)d_in[5];  // [8]
  float* out = (float*)d_out;                // [2,1024,8]

  float* g1 = (float*)d_ws;                  // [2048, 64]
  float* h  = g1 + (long)NB * NN * 64;       // [2048, 64]
  float* g2 = h + (long)NB * NN * 64;        // [2048, 8]

  const int rowTiles = NB * NN / 16;  // 128

  // layer 1: g1 = x @ W1^T
  gemm_xwt_wmma<128, 64, 64><<<rowTiles, 32, 0, stream>>>(x, W1, g1);
  // layer 1 attention (8 heads) + ELU -> h
  gatv2_attn<64, true><<<rowTiles, 256, 0, stream>>>(g1, adj, a1, h);
  // layer 2: g2 = h @ W2^T (8 valid cols, padded tile)
  gemm_xwt_wmma<64, 16, 8><<<rowTiles, 32, 0, stream>>>(h, W2, g2);
  // layer 2 attention (1 head), mean over 1 head == identity -> out
  gatv2_attn<8, false><<<rowTiles, 32, 0, stream>>>(g2, adj, a2, out);
}